// DistributeInputLayer_34222299414942
// MI455X (gfx1250) — compile-verified
//
#include <hip/hip_runtime.h>
#include <hip/hip_bf16.h>
#include <stdint.h>

// im2col unfold: out[b, w, f, c] = x[b, w+f, c]
//   x:   [128, 4096, 4]  f32   (each row of 4 channels = one 16B float4)
//   out: [128, 4065, 32, 4] f32
//
// Pure data movement (0 FLOPs): write-bandwidth bound (~266 MB of stores).
// Strategy: stage each block's 159 input rows (2544 B) into LDS via CDNA5
// async global->LDS b128 loads (ASYNCcnt), then emit coalesced 128-bit
// non-temporal stores fed by bank-conflict-free ds_load_b128.

#define FILTER_WIDTH 32
#define SEQ_LEN      4096
#define BATCH        128
#define N_WIN        (SEQ_LEN - FILTER_WIDTH + 1)   // 4065
#define WPB          128                             // windows per block
#define THREADS      256                             // 8 waves (wave32)
#define ROWS_MAX     (WPB + FILTER_WIDTH - 1)        // 159 staged rows
#define F4_PER_THREAD ((WPB * FILTER_WIDTH) / THREADS) // 16 stores/thread

typedef float float4v __attribute__((ext_vector_type(4)));

__global__ __launch_bounds__(THREADS)
void DistributeInputLayer_34222299414942_kernel(const float4v* __restrict__ x,
                                                float4v* __restrict__ out) {
    __shared__ float4v smem[ROWS_MAX + 1];   // +1 pad, 2560 B total

    const int tid = threadIdx.x;
    const int b   = blockIdx.y;
    const int w0  = blockIdx.x * WPB;

    // ---- Stage input rows [w0, w0+rows) of batch b into LDS --------------
    // GVS async copy: global addr = SGPR64 base + VGPR byte offset.
    const float4v* xrow = x + (size_t)b * SEQ_LEN + w0;     // uniform -> SGPR pair
    const int rows = min(ROWS_MAX, SEQ_LEN - w0);           // last tile clamps

    if (tid < rows) {
        unsigned lds_addr = (unsigned)(uintptr_t)(&smem[tid]); // low 32b = LDS byte offset
        unsigned goff     = (unsigned)(tid * (int)sizeof(float4v));
        asm volatile("global_load_async_to_lds_b128 %0, %1, %2"
                     :: "v"(lds_addr), "v"(goff), "s"(xrow)
                     : "memory");
    }
    // Drain this wave's async copies, then workgroup barrier so every wave
    // sees the fully staged tile.
    asm volatile("s_wait_asynccnt 0" ::: "memory");
    __syncthreads();

    // ---- Emit output: out[b, w0+wl, f, :] = smem[wl + f] -----------------
    // o = wl*32 + f; consecutive tid -> consecutive o -> perfectly coalesced
    // 16B/lane global stores. LDS reads hit all 64 banks evenly.
    float4v* outb = out + ((size_t)b * N_WIN + w0) * FILTER_WIDTH;

    if (w0 + WPB <= N_WIN) {
        // Fast path: full tile, branch-free inner loop.
        #pragma unroll
        for (int k = 0; k < F4_PER_THREAD; ++k) {
            const int o  = tid + k * THREADS;
            const int wl = o >> 5;        // window within tile
            const int f  = o & 31;        // filter tap
            __builtin_nontemporal_store(smem[wl + f], &outb[o]);
        }
    } else {
        // Tail tile (97 windows): guard window index.
        #pragma unroll
        for (int k = 0; k < F4_PER_THREAD; ++k) {
            const int o  = tid + k * THREADS;
            const int wl = o >> 5;
            const int f  = o & 31;
            if (w0 + wl < N_WIN) {
                __builtin_nontemporal_store(smem[wl + f], &outb[o]);
            }
        }
    }
}

extern "C" void kernel_launch(void* const* d_in, const int* in_sizes, int n_in,
                              void* d_out, int out_size, void* d_ws, size_t ws_size,
                              hipStream_t stream) {
    const float4v* x   = (const float4v*)d_in[0];
    float4v*       out = (float4v*)d_out;

    dim3 grid((N_WIN + WPB - 1) / WPB, BATCH);   // 32 x 128 = 4096 blocks
    DistributeInputLayer_34222299414942_kernel<<<grid, THREADS, 0, stream>>>(x, out);
}